// BKT_47785806135758
// MI455X (gfx1250) — compile-verified
//
#include <hip/hip_runtime.h>
#include <stdint.h>

// ---------------------------------------------------------------------------
// BKT scan for MI455X (gfx1250).
//   responses : (B, L) int32 (0/1)
//   outputs   : pred (B, L-1) f32  ||  r_shft (B, L-1) f32   (concatenated)
//
// Roofline: ~100 MB traffic @ 23.3 TB/s => 4.3 us floor; ~67 MFLOP (AI ~ 0.7
// flop/byte) => memory/latency bound, WMMA not applicable. Parallelism is
// fixed at 8192 rows = 256 wave32, so the per-step serial chain dominates:
//   - input staging: CDNA5 Tensor Data Mover (tensor_load_to_lds), double
//     buffered, TENSORcnt split-waits, LDS rows padded to 33 DW (bank-free).
//   - recurrence: Mobius map carried in homogeneous (u,v) form ->
//     chain = mul+fma (~8 cyc/step); the division is off-chain via
//     v_rcp_f32 (1 ulp). Renormalize every 16 steps (factors in [0.1,0.9]
//     => |v| >= ~1e-16 between renorms, no fp32 underflow).
// ---------------------------------------------------------------------------

#define B_ROWS        8192
#define L_COLS        1024
#define ROWS_PER_BLK  128          // 4 wave32 per block
#define TILE_W        32           // int32 columns per TDM tile (128 B / row)
#define ROW_STRIDE    33           // padded LDS row stride (33 coprime 64 banks)
#define NTILES        (L_COLS / TILE_W)
#define RENORM        16           // homogeneous renormalization period

typedef unsigned int v4u __attribute__((ext_vector_type(4)));
typedef int          v8i __attribute__((ext_vector_type(8)));
typedef int          v4i __attribute__((ext_vector_type(4)));

// Issue one TDM load: 2-D tile [ROWS_PER_BLK x TILE_W int32] of the
// (B_ROWS x L_COLS) int32 tensor into LDS at lds_off, 1-DWORD pad per 32 DW.
// (Descriptor encoding verified in round-1 asm: tensor_load_to_lds with the
//  exact D# words in SGPRs, groups 2/3 NULL.)
__device__ static inline void tdm_issue_tile(uint32_t lds_off, const int* tile_gptr) {
  uint64_t ga = (uint64_t)(uintptr_t)tile_gptr;

  v4u g0;
  g0[0] = 1u;                                   // count=1, user mode, no gather
  g0[1] = lds_off;                              // LDS byte address
  g0[2] = (uint32_t)(ga & 0xFFFFFFFFu);         // global_addr[31:0]
  g0[3] = ((uint32_t)(ga >> 32) & 0x01FFFFFFu)  // global_addr[56:32]
          | 0x80000000u;                        // type=2 ("image")

  v8i g1;
  g1[0] = (2 << 16)        // data_size = 4 bytes
        | (1 << 20)        // pad_enable
        | (4 << 22)        // pad_interval: code 4 -> every 32 DWORDs
        | (0 << 25);       // pad_amount:   code 0 -> 1 DWORD
  g1[1] = (int)(((uint32_t)L_COLS & 0xFFFFu) << 16);  // tensor_dim0[15:0]
  g1[2] = (int)(((uint32_t)B_ROWS & 0xFFFFu) << 16);  // tensor_dim1[15:0]
  g1[3] = (int)((uint32_t)TILE_W << 16);              // tile_dim0
  g1[4] = ROWS_PER_BLK;                               // tile_dim1 (tile_dim2=0)
  g1[5] = L_COLS;                                     // tensor_dim0_stride
  g1[6] = 0;
  g1[7] = 0;

  v4i z4 = {0, 0, 0, 0};   // groups 2/3 -> NULL (2-D tile)
#if defined(__clang_major__) && (__clang_major__ >= 23)
  v8i z8 = {0, 0, 0, 0, 0, 0, 0, 0};
  __builtin_amdgcn_tensor_load_to_lds(g0, g1, z4, z4, z8, 0);
#else
  __builtin_amdgcn_tensor_load_to_lds(g0, g1, z4, z4, 0);
#endif
}

__global__ __launch_bounds__(ROWS_PER_BLK)
void bkt_scan_kernel(const int* __restrict__ responses,
                     const float* __restrict__ p_slip,
                     const float* __restrict__ p_guess,
                     const float* __restrict__ p_train,
                     const float* __restrict__ p_learn,
                     float* __restrict__ out) {
  __shared__ int tile[2][ROWS_PER_BLK * ROW_STRIDE];

  const int tid = threadIdx.x;
  const int row = blockIdx.x * ROWS_PER_BLK + tid;

  const float s = *p_slip;
  const float g = *p_guess;
  const float p = *p_train;

  // Mobius coefficients: K' = (a*K + b) / (c*K + d) == p + (1-p)*k_select.
  //   r==1: k = K(1-s) / (K(1-s-g) + g)
  //   r==0: k = K s     / (K(s+g-1) + (1-g))
  const float omp = 1.0f - p;
  const float c1f = 1.0f - s - g;          // beta  (r=1)
  const float d1f = g;                     // gamma (r=1)
  const float a1f = p * c1f + omp * (1.0f - s);
  const float b1f = p * d1f;
  const float c0f = s + g - 1.0f;          // beta  (r=0)
  const float d0f = 1.0f - g;              // gamma (r=0)
  const float a0f = p * c0f + omp * s;
  const float b0f = p * d0f;

  // Homogeneous state: K = u/v. K0 = learn_p.
  float u = *p_learn;
  float v = 1.0f;

  float* __restrict__ pred  = out + (size_t)row * (L_COLS - 1);
  float* __restrict__ rshft = out + (size_t)B_ROWS * (L_COLS - 1)
                                  + (size_t)row * (L_COLS - 1);
  const int* rowbase = responses + (size_t)blockIdx.x * ROWS_PER_BLK * L_COLS;

  // Prologue: DMA tile 0 into buffer 0 (one TDM op per block, wave 0 issues).
  if (tid == 0) {
    tdm_issue_tile((uint32_t)(uintptr_t)&tile[0][0], rowbase);
  }

  for (int tI = 0; tI < NTILES; ++tI) {
    const int cur = tI & 1;

    if (tid == 0) {
      if (tI + 1 < NTILES) {
        // Prefetch next tile; previous iteration's trailing barrier proved
        // buffer cur^1 is free. In-order TDM: <=1 outstanding => tile tI done.
        tdm_issue_tile((uint32_t)(uintptr_t)&tile[cur ^ 1][0],
                       rowbase + (tI + 1) * TILE_W);
        __builtin_amdgcn_s_wait_tensorcnt(1);
      } else {
        __builtin_amdgcn_s_wait_tensorcnt(0);
      }
    }
    __syncthreads();                      // tile tI visible to all 4 waves

    const int* __restrict__ lrow = &tile[cur][tid * ROW_STRIDE];
    const int  c0 = tI * TILE_W;

#pragma unroll
    for (int j = 0; j < TILE_W; ++j) {
      const int c    = c0 + j;            // global column in responses
      const int resp = lrow[j];           // bank-conflict-free (stride 33)

      if (c >= 1) {                       // r_shft = float(responses[:, 1:])
        rshft[c - 1] = (float)resp;
      }
      if (c <= L_COLS - 2) {              // scan consumes r_in = responses[:, :-1]
        const int  m    = (resp > -1) ? resp : 0;   // masked_r (no-op for 0/1)
        const bool corr = (m == 1);
        const float a = corr ? a1f : a0f;
        const float b = corr ? b1f : b0f;
        const float cc = corr ? c1f : c0f;
        const float d = corr ? d1f : d0f;

        // Critical chain: two independent mul+fma pairs (~8 cycles).
        const float un = fmaf(a, u, b * v);
        const float vn = fmaf(cc, u, d * v);

        // Off-chain output: Kt = un/vn via 1-ulp v_rcp_f32.
        const float rv = __builtin_amdgcn_rcpf(vn);
        pred[c] = un * rv;

        if ((j & (RENORM - 1)) == (RENORM - 1)) {
          u = un * rv;                    // renormalize: v -> 1 (reuses rv)
          v = 1.0f;
        } else {
          u = un;
          v = vn;
        }
      }
    }
    __syncthreads();                      // done reading buf `cur` before reuse
  }
}

extern "C" void kernel_launch(void* const* d_in, const int* in_sizes, int n_in,
                              void* d_out, int out_size, void* d_ws, size_t ws_size,
                              hipStream_t stream) {
  (void)in_sizes; (void)n_in; (void)out_size; (void)d_ws; (void)ws_size;
  const int*   responses = (const int*)d_in[0];
  const float* slip      = (const float*)d_in[1];
  const float* guess     = (const float*)d_in[2];
  const float* train_p   = (const float*)d_in[3];
  const float* learn_p   = (const float*)d_in[4];
  float*       out       = (float*)d_out;

  dim3 grid(B_ROWS / ROWS_PER_BLK);      // 64 blocks
  dim3 block(ROWS_PER_BLK);              // 128 threads = 4 wave32
  bkt_scan_kernel<<<grid, block, 0, stream>>>(responses, slip, guess,
                                              train_p, learn_p, out);
}